// LSTM_3040836845939
// MI455X (gfx1250) — compile-verified
//
#include <hip/hip_runtime.h>
#include <hip/hip_bf16.h>
#include <math.h>

// Problem constants (match reference)
#define TLEN 262144
#define HD   90
#define NG   360           // 4*H gate outputs
#define GS   368           // padded gate stride: 23 tiles of 16

typedef __attribute__((ext_vector_type(2))) float v2f;
typedef __attribute__((ext_vector_type(8))) float v8f;

// ---------------------------------------------------------------------------
// Packing kernels: move all zero-padding / transposition OUT of the WMMA hot
// loop so the GEMM inner loop is pure b64 loads + v_wmma with EXEC all-1s.
// ---------------------------------------------------------------------------

// Xp[t*KP + k] = (k < Din) ? X[t*Din + k] : 0
__global__ void pack_x(const float* __restrict__ X, int Din, int KP, int total,
                       float* __restrict__ Xp)
{
    const int i = blockIdx.x * blockDim.x + threadIdx.x;
    if (i >= total) return;
    const int k = i % KP;
    const int t = i / KP;
    Xp[i] = (k < Din) ? X[(size_t)t * Din + k] : 0.0f;
}

// Wp packed in per-lane WMMA-B fragment order: [kc][hi][n][e], e in {0,1}.
// Lane (hi, l16) of chunk kc reads float2 { B[4kc+2hi][n], B[4kc+2hi+1][n] }
// where B[k][n] = W[n][k].
__global__ void pack_w(const float* __restrict__ W, int Din, int NC,
                       float* __restrict__ Wp)
{
    const int i = blockIdx.x * blockDim.x + threadIdx.x;
    const int total = NC * 2 * 368 * 2;
    if (i >= total) return;
    const int e  = i & 1;
    const int n  = (i >> 1) % 368;
    const int r  = (i >> 1) / 368;
    const int hi = r & 1;
    const int kc = r >> 1;
    const int k  = 4 * kc + 2 * hi + e;
    Wp[i] = (n < NG && k < Din) ? W[(size_t)n * Din + k] : 0.0f;
}

// bsum[n] = b_ih[n] + b_hh[n], zero-padded to 368
__global__ void pack_bias(const float* __restrict__ bih,
                          const float* __restrict__ bhh,
                          float* __restrict__ bsum)
{
    const int i = blockIdx.x * blockDim.x + threadIdx.x;
    if (i >= 368) return;
    bsum[i] = (i < NG) ? (bih[i] + bhh[i]) : 0.0f;
}

// ---------------------------------------------------------------------------
// Kernel 1: G[t, n] = sum_k Xp[t,k] * B[k,n] + bsum[n]  via fp32 WMMA 16x16x4.
// (Exact fp32 -- this GEMM is bandwidth-bound, low precision buys nothing.)
// One wave per 16x16 output tile; K-loop fully unrolled at compile time.
//
// VGPR layouts per CDNA5 ISA 7.12.2:
//   A (16x4 f32): lanes 0-15 -> M=lane, {K0,K1}; lanes 16-31 -> M=lane-16, {K2,K3}
//   B (4x16 f32): lanes 0-15 -> N=lane, {K0,K1}; lanes 16-31 -> N=lane-16, {K2,K3}
//   C/D (16x16 f32): VGPR r: lanes 0-15 -> M=r, lanes 16-31 -> M=8+r; N=lane&15
// ---------------------------------------------------------------------------
template <int KP>
__global__ void __launch_bounds__(256)
xproj_wmma(const float* __restrict__ Xp,   // [T, KP] padded
           const float* __restrict__ Wp,   // packed B fragments
           const float* __restrict__ bsum, // [368]
           float* __restrict__ G)          // [T, GS]
{
    const int lane  = threadIdx.x & 31;
    const int wave  = threadIdx.x >> 5;                 // 0..7
    const int mtile = blockIdx.x;                       // T/16 tiles
    const int ntile = blockIdx.y * 8 + wave;            // 24 slots, 23 used
    if (ntile >= 23) return;                            // wave-uniform branch

    const int hi  = lane >> 4;                          // 0 or 1
    const int l16 = lane & 15;
    const int m   = mtile * 16 + l16;                   // A row owned by this lane
    const int n   = ntile * 16 + l16;                   // B/C column owned by this lane

    const float bias = bsum[n];
    v8f acc;
#pragma unroll
    for (int r = 0; r < 8; ++r) acc[r] = bias;

    const float* ap = Xp + (size_t)m * KP + 2 * hi;     // 8B-aligned float2 per chunk
    const float* bp = Wp + ((size_t)hi * 368 + n) * 2;  // coalesced float2 per chunk
    constexpr int NC = KP / 4;
#pragma unroll
    for (int kc = 0; kc < NC; ++kc) {
        const v2f a = *(const v2f*)(ap + 4 * kc);
        const v2f b = *(const v2f*)(bp + (size_t)kc * (2 * 368 * 2));
        acc = __builtin_amdgcn_wmma_f32_16x16x4_f32(
                  /*neg_a=*/false, a, /*neg_b=*/false, b,
                  /*c_mod=*/(short)0, acc, /*reuse_a=*/false, /*reuse_b=*/false);
    }

#pragma unroll
    for (int r = 0; r < 8; ++r) {
        const int mm = mtile * 16 + hi * 8 + r;
        G[(size_t)mm * GS + ntile * 16 + l16] = acc[r]; // padded stride: always in range
    }
}

// ---------------------------------------------------------------------------
// Kernel 2: sequential LSTM scan for one layer. Single persistent workgroup,
// 384 threads (12 wave32). Lane j (j<360) owns gate output j: W_hh[j][:] lives
// in 90 VGPRs, h_{t-1} is broadcast from LDS. G[t+1] is software-pipelined and
// G[t+2] prefetched so HBM latency hides under the FMA chain.
// Gate order (reference split): i=[0,90) f=[90,180) g=[180,270) o=[270,360)
// ---------------------------------------------------------------------------
__global__ void __launch_bounds__(384, 1)
lstm_scan(const float* __restrict__ G,     // [T, GS]
          const float* __restrict__ Whh,   // [360, 90] this layer
          const float* __restrict__ h0,    // [90]
          const float* __restrict__ c0,    // [90]
          float* __restrict__ hs)          // [T, 90] out
{
    __shared__ float h_lds[HD];
    __shared__ float g_lds[NG];
    const int j = threadIdx.x;             // 0..383 (360..383 idle helpers)

    float wreg[HD];
    if (j < NG) {
#pragma unroll
        for (int k = 0; k < HD; ++k) wreg[k] = Whh[(size_t)j * HD + k];
    }
    float c = 0.0f;
    if (j < HD) { h_lds[j] = h0[j]; c = c0[j]; }
    __syncthreads();

    float gcur = (j < NG) ? G[j] : 0.0f;
    for (int t = 0; t < TLEN; ++t) {
        float gnext = 0.0f;
        if (j < NG && t + 1 < TLEN) {
            gnext = G[(size_t)(t + 1) * GS + j];
            if (t + 2 < TLEN) __builtin_prefetch(&G[(size_t)(t + 2) * GS + j], 0, 0);
        }
        if (j < NG) {
            // matvec: acc = gcur + sum_k Whh[j][k] * h[k]; two accumulators
            // shorten the dependent-FMA chain.
            float a0 = gcur, a1 = 0.0f;
#pragma unroll
            for (int k = 0; k < HD; k += 2) {
                a0 += wreg[k]     * h_lds[k];
                a1 += wreg[k + 1] * h_lds[k + 1];
            }
            const float pre  = a0 + a1;
            const int   gate = j / HD;
            const float act  = (gate == 2) ? tanhf(pre)
                                           : 1.0f / (1.0f + __expf(-pre));
            g_lds[j] = act;
        }
        __syncthreads();
        if (j < HD) {
            c = g_lds[HD + j] * c + g_lds[j] * g_lds[2 * HD + j];   // f*c + i*g
            const float hn = g_lds[3 * HD + j] * tanhf(c);          // o*tanh(c)
            h_lds[j] = hn;
            hs[(size_t)t * HD + j] = hn;
        }
        __syncthreads();
        gcur = gnext;
    }
}

// ---------------------------------------------------------------------------
// Kernel 3: head -- logits = h @ fc_w^T + fc_b, softmax, argmax.
// probs -> out[0 .. 2T), zv (int32) -> out[2T .. 3T) reinterpreted.
// ---------------------------------------------------------------------------
__global__ void head_kernel(const float* __restrict__ hs,
                            const float* __restrict__ fcw,  // [2,90]
                            const float* __restrict__ fcb,  // [2]
                            float* __restrict__ out)
{
    __shared__ float w[2 * HD];
    __shared__ float b[2];
    if (threadIdx.x < 2 * HD) w[threadIdx.x] = fcw[threadIdx.x];
    if (threadIdx.x < 2)      b[threadIdx.x] = fcb[threadIdx.x];
    __syncthreads();

    const int t = blockIdx.x * blockDim.x + threadIdx.x;
    if (t >= TLEN) return;

    float d0 = b[0], d1 = b[1];
#pragma unroll 6
    for (int k = 0; k < HD; ++k) {
        const float hv = hs[(size_t)t * HD + k];
        d0 += w[k] * hv;
        d1 += w[HD + k] * hv;
    }
    const float mx  = fmaxf(d0, d1);
    const float e0  = __expf(d0 - mx);
    const float e1  = __expf(d1 - mx);
    const float inv = 1.0f / (e0 + e1);
    out[2 * t]     = e0 * inv;
    out[2 * t + 1] = e1 * inv;
    ((int*)out)[2 * TLEN + t] = (d1 > d0) ? 1 : 0;  // argmax, ties -> 0 like jnp.argmax
}

// ---------------------------------------------------------------------------
extern "C" void kernel_launch(void* const* d_in, const int* in_sizes, int n_in,
                              void* d_out, int out_size, void* d_ws, size_t ws_size,
                              hipStream_t stream)
{
    (void)in_sizes; (void)n_in; (void)out_size; (void)ws_size;

    const float* x    = (const float*)d_in[0];   // [T,6]
    const float* h0   = (const float*)d_in[1];   // [4,90]
    const float* c0   = (const float*)d_in[2];   // [4,90]
    const float* wih[4] = { (const float*)d_in[3], (const float*)d_in[4],
                            (const float*)d_in[5], (const float*)d_in[6] };
    const float* whh  = (const float*)d_in[7];   // [4,360,90]
    const float* bih  = (const float*)d_in[8];   // [4,360]
    const float* bhh  = (const float*)d_in[9];   // [4,360]
    const float* fcw  = (const float*)d_in[10];  // [2,90]
    const float* fcb  = (const float*)d_in[11];  // [2]

    // Workspace: G [T,368] | hA [T,90] | hB [T,90] | Xp [T,92] | Wp | bsum
    float* G    = (float*)d_ws;
    float* hA   = G    + (size_t)TLEN * GS;
    float* hB   = hA   + (size_t)TLEN * HD;
    float* Xp   = hB   + (size_t)TLEN * HD;
    float* Wp   = Xp   + (size_t)TLEN * 92;
    float* bsum = Wp   + (size_t)23 * 2 * 368 * 2;
    float* bufs[2] = { hA, hB };

    const dim3 ggrid(TLEN / 16, 3);   // 16384 M-tiles x 3 (8 waves -> 23 N-tiles)
    const float* src = x;
    for (int l = 0; l < 4; ++l) {
        const int din = (l == 0) ? 6 : HD;
        const int kp  = (l == 0) ? 8 : 92;
        const int nc  = kp / 4;

        {   // pack A (zero-padded), B (fragment order), bias
            const int totalx = TLEN * kp;
            pack_x<<<(totalx + 255) / 256, 256, 0, stream>>>(src, din, kp, totalx, Xp);
            const int totalw = nc * 2 * 368 * 2;
            pack_w<<<(totalw + 255) / 256, 256, 0, stream>>>(wih[l], din, nc, Wp);
            pack_bias<<<2, 192, 0, stream>>>(bih + (size_t)l * NG,
                                             bhh + (size_t)l * NG, bsum);
        }

        if (l == 0) xproj_wmma<8><<<ggrid, 256, 0, stream>>>(Xp, Wp, bsum, G);
        else        xproj_wmma<92><<<ggrid, 256, 0, stream>>>(Xp, Wp, bsum, G);

        float* hout = bufs[l & 1];
        lstm_scan<<<1, 384, 0, stream>>>(G, whh + (size_t)l * NG * HD,
                                         h0 + (size_t)l * HD, c0 + (size_t)l * HD,
                                         hout);
        src = hout;
    }
    head_kernel<<<(TLEN + 255) / 256, 256, 0, stream>>>(src, fcw, fcb, (float*)d_out);
}